// GraphClassifier_41781441855681
// MI455X (gfx1250) — compile-verified
//
#include <hip/hip_runtime.h>
#include <hip/hip_bf16.h>

#define NN   100000   // nodes
#define NE   1600000  // edges
#define NG   1000     // graphs
#define DH   64       // feature dim (in & hidden)
#define DMID 32
#define DOUT 8

typedef __attribute__((ext_vector_type(2))) float v2f;
typedef __attribute__((ext_vector_type(8))) float v8f;

// ---------------- small utility kernels ----------------

__global__ void fill_kernel(float* __restrict__ p, float v, int n) {
    int i = blockIdx.x * blockDim.x + threadIdx.x;
    if (i < n) p[i] = v;
}

// deg starts at 1.0 (self loop); count incoming edges per dst
__global__ void deg_count_kernel(const int* __restrict__ dst, float* __restrict__ deg, int e) {
    int i = blockIdx.x * blockDim.x + threadIdx.x;
    if (i < e) atomicAdd(&deg[dst[i]], 1.0f);
}

// in-place deg -> dinv = rsqrt(deg)  (deg >= 1 always due to self loops)
__global__ void rsqrt_kernel(float* __restrict__ p, int n) {
    int i = blockIdx.x * blockDim.x + threadIdx.x;
    if (i < n) p[i] = rsqrtf(p[i]);
}

// ---------------- WMMA FP32 GEMM: Y[n,64] = X[n,64] @ W[64,64] ----------------
// One wave computes a 16-row x 64-col slab using V_WMMA_F32_16X16X4_F32.
// A frag (16x4 f32, 2 VGPRs): lane l -> M = l%16, K pair base = 2*(l/16)
// B frag (4x16 f32, 2 VGPRs): lane l -> N = l%16, K pair base = 2*(l/16)
// C/D (16x16 f32, 8 VGPRs):   lane l -> N = l%16, M = vgpr + 8*(l/16)
__global__ void gemm_wmma_f32(const float* __restrict__ X, const float* __restrict__ W,
                              float* __restrict__ Y, int nrows) {
    const int gtid = blockIdx.x * blockDim.x + threadIdx.x;
    const int wave = gtid >> 5;
    const int lane = threadIdx.x & 31;
    const int row0 = wave * 16;
    if (row0 >= nrows) return;            // wave-uniform: EXEC stays all-ones

    const int m  = lane & 15;
    const int kb = (lane >> 4) * 2;       // 0 or 2

    v8f acc0 = {}, acc1 = {}, acc2 = {}, acc3 = {};

    #pragma unroll 4
    for (int k0 = 0; k0 < DH; k0 += 4) {
        // A fragment: X[row0+m][k0+kb .. +1]  (64-bit contiguous load per lane)
        const float* ap = X + (size_t)(row0 + m) * DH + (k0 + kb);
        v2f a; a.x = ap[0]; a.y = ap[1];

        // B fragments for 4 column tiles: W[k0+kb][col0+m], W[k0+kb+1][col0+m]
        const float* bp = W + (size_t)(k0 + kb) * DH + m;
        v2f b0; b0.x = bp[0];  b0.y = bp[DH];
        v2f b1; b1.x = bp[16]; b1.y = bp[DH + 16];
        v2f b2; b2.x = bp[32]; b2.y = bp[DH + 32];
        v2f b3; b3.x = bp[48]; b3.y = bp[DH + 48];

        acc0 = __builtin_amdgcn_wmma_f32_16x16x4_f32(false, a, false, b0, (short)0, acc0, false, false);
        acc1 = __builtin_amdgcn_wmma_f32_16x16x4_f32(false, a, false, b1, (short)0, acc1, false, false);
        acc2 = __builtin_amdgcn_wmma_f32_16x16x4_f32(false, a, false, b2, (short)0, acc2, false, false);
        acc3 = __builtin_amdgcn_wmma_f32_16x16x4_f32(false, a, false, b3, (short)0, acc3, false, false);
    }

    const int n     = lane & 15;
    const int mBase = (lane >> 4) * 8;
    #pragma unroll
    for (int i = 0; i < 8; ++i) {
        float* yr = Y + (size_t)(row0 + mBase + i) * DH + n;
        yr[0]  = acc0[i];
        yr[16] = acc1[i];
        yr[32] = acc2[i];
        yr[48] = acc3[i];
    }
}

// ---------------- edge scatter: agg[dst] += T[src] * dinv[src]*dinv[dst] ----------------
// one thread per (edge, 4-dim chunk); 16 consecutive threads cover one edge's 64-float row
__global__ void scatter_edges_kernel(const float* __restrict__ T, const int* __restrict__ src,
                                     const int* __restrict__ dst, const float* __restrict__ dinv,
                                     float* __restrict__ agg, int e) {
    int idx = blockIdx.x * blockDim.x + threadIdx.x;
    if (idx >= e * 16) return;
    const int ei = idx >> 4;
    const int c  = (idx & 15) * 4;
    const int s  = src[ei];
    const int d  = dst[ei];
    const float nrm = dinv[s] * dinv[d];
    const float4 v = *(const float4*)(T + (size_t)s * DH + c);
    float* o = agg + (size_t)d * DH + c;
    atomicAdd(o + 0, v.x * nrm);
    atomicAdd(o + 1, v.y * nrm);
    atomicAdd(o + 2, v.z * nrm);
    atomicAdd(o + 3, v.w * nrm);
}

// add self-loop term + bias, optional relu (in place on agg)
__global__ void finish_conv_kernel(float* __restrict__ agg, const float* __restrict__ T,
                                   const float* __restrict__ dinv, const float* __restrict__ bias,
                                   int total, int do_relu) {
    int idx = blockIdx.x * blockDim.x + threadIdx.x;
    if (idx >= total) return;
    const int node = idx >> 6;      // / DH
    const int d    = idx & 63;      // % DH
    const float di = dinv[node];
    float v = agg[idx] + T[idx] * di * di + bias[d];
    agg[idx] = do_relu ? fmaxf(v, 0.0f) : v;
}

// ---------------- global mean pool (accumulate) ----------------
__global__ void pool_accum_kernel(const float* __restrict__ h, const int* __restrict__ batch,
                                  float* __restrict__ pooled, float* __restrict__ counts, int total) {
    int idx = blockIdx.x * blockDim.x + threadIdx.x;
    if (idx >= total) return;
    const int node = idx >> 6;
    const int d    = idx & 63;
    const int g    = batch[node];
    atomicAdd(&pooled[(size_t)g * DH + d], h[idx]);
    if (d == 0) atomicAdd(&counts[g], 1.0f);
}

// ---------------- MLP head: out = relu(pooled/cnt @ lw1 + lb1) @ lw2 + lb2 ----------------
__global__ void head_kernel(const float* __restrict__ pooled, const float* __restrict__ counts,
                            const float* __restrict__ lw1, const float* __restrict__ lb1,
                            const float* __restrict__ lw2, const float* __restrict__ lb2,
                            float* __restrict__ out) {
    int g = blockIdx.x * blockDim.x + threadIdx.x;
    if (g >= NG) return;
    const float inv = 1.0f / fmaxf(counts[g], 1.0f);
    float p[DH];
    #pragma unroll
    for (int k = 0; k < DH; ++k) p[k] = pooled[(size_t)g * DH + k] * inv;
    float z[DMID];
    for (int j = 0; j < DMID; ++j) {
        float acc = lb1[j];
        #pragma unroll
        for (int k = 0; k < DH; ++k) acc = fmaf(p[k], lw1[k * DMID + j], acc);
        z[j] = fmaxf(acc, 0.0f);
    }
    for (int o = 0; o < DOUT; ++o) {
        float acc = lb2[o];
        #pragma unroll
        for (int k = 0; k < DMID; ++k) acc = fmaf(z[k], lw2[k * DOUT + o], acc);
        out[(size_t)g * DOUT + o] = acc;
    }
}

// ---------------- host orchestration ----------------

extern "C" void kernel_launch(void* const* d_in, const int* in_sizes, int n_in,
                              void* d_out, int out_size, void* d_ws, size_t ws_size,
                              hipStream_t stream) {
    const float* x     = (const float*)d_in[0];
    const int*   ei    = (const int*)d_in[1];
    const int*   src   = ei;             // edge_index[0]
    const int*   dst   = ei + NE;        // edge_index[1]
    const int*   batch = (const int*)d_in[2];
    const float* W1    = (const float*)d_in[3];
    const float* b1    = (const float*)d_in[4];
    const float* W2    = (const float*)d_in[5];
    const float* b2    = (const float*)d_in[6];
    const float* lw1   = (const float*)d_in[7];
    const float* lb1   = (const float*)d_in[8];
    const float* lw2   = (const float*)d_in[9];
    const float* lb2   = (const float*)d_in[10];
    float* out = (float*)d_out;

    // workspace carve-up (floats), 256-float aligned
    float* ws     = (float*)d_ws;
    float* dinv   = ws;                                   // NN
    float* bufA   = dinv + ((NN + 255) & ~255);           // NN*DH   (t1 / t2)
    float* bufB   = bufA + (size_t)NN * DH;               // NN*DH   (agg/h1, agg/h2)
    float* pooled = bufB + (size_t)NN * DH;               // NG*DH
    float* counts = pooled + (size_t)NG * DH;             // NG

    const int BT = 256;
    const int gemmWaves  = NN / 16;                        // 6250 exactly
    const int gemmBlocks = (gemmWaves * 32 + BT - 1) / BT;
    const int nodeElems  = NN * DH;
    const int edgeThreads = NE * 16;

    // degree -> dinv
    fill_kernel<<<(NN + BT - 1) / BT, BT, 0, stream>>>(dinv, 1.0f, NN);
    deg_count_kernel<<<(NE + BT - 1) / BT, BT, 0, stream>>>(dst, dinv, NE);
    rsqrt_kernel<<<(NN + BT - 1) / BT, BT, 0, stream>>>(dinv, NN);

    // ---- layer 1 ----
    gemm_wmma_f32<<<gemmBlocks, BT, 0, stream>>>(x, W1, bufA, NN);                       // t1
    fill_kernel<<<(nodeElems + BT - 1) / BT, BT, 0, stream>>>(bufB, 0.0f, nodeElems);
    scatter_edges_kernel<<<(edgeThreads + BT - 1) / BT, BT, 0, stream>>>(bufA, src, dst, dinv, bufB, NE);
    finish_conv_kernel<<<(nodeElems + BT - 1) / BT, BT, 0, stream>>>(bufB, bufA, dinv, b1, nodeElems, 1); // h1

    // ---- layer 2 ----
    gemm_wmma_f32<<<gemmBlocks, BT, 0, stream>>>(bufB, W2, bufA, NN);                    // t2
    fill_kernel<<<(nodeElems + BT - 1) / BT, BT, 0, stream>>>(bufB, 0.0f, nodeElems);
    scatter_edges_kernel<<<(edgeThreads + BT - 1) / BT, BT, 0, stream>>>(bufA, src, dst, dinv, bufB, NE);
    finish_conv_kernel<<<(nodeElems + BT - 1) / BT, BT, 0, stream>>>(bufB, bufA, dinv, b2, nodeElems, 0); // h2

    // ---- pool + head ----
    fill_kernel<<<(NG * DH + NG + BT - 1) / BT, BT, 0, stream>>>(pooled, 0.0f, NG * DH + NG); // pooled+counts contiguous
    pool_accum_kernel<<<(nodeElems + BT - 1) / BT, BT, 0, stream>>>(bufB, batch, pooled, counts, nodeElems);
    head_kernel<<<(NG + 127) / 128, 128, 0, stream>>>(pooled, counts, lw1, lb1, lw2, lb2, out);
}